// HCNF_85409719648823
// MI455X (gfx1250) — compile-verified
//
#include <hip/hip_runtime.h>

// Hyperbolic CNF forward: B=4096 samples, D=32, H=128, 16 RK4 steps.
// One wave32 per sample, 8 waves per block sharing an LDS copy of W1/W2/W3.
// Primal MLP = VALU matvecs from LDS; exact-divergence JVP (two 32x128x128
// GEMMs per RHS eval) = V_WMMA_F32_16X16X4_F32 with A staged in per-wave LDS
// and B read via ds_load_2addr from the shared weight copy.

#define B_N    4096
#define D_N    32
#define H_N    128
#define NSTEPS 16
#define DT_F   (1.0f / 16.0f)
#define US     132          // padded LDS row stride for U (floats)
#define WS     136          // padded LDS row stride for W1/W2 (floats)
#define WAVES  8

typedef __attribute__((ext_vector_type(2))) float v2f;
typedef __attribute__((ext_vector_type(4))) float v4f;
typedef __attribute__((ext_vector_type(8))) float v8f;

#if defined(__has_builtin)
#if __has_builtin(__builtin_amdgcn_tanhf)
#define TANHF(x) __builtin_amdgcn_tanhf(x)
#else
#define TANHF(x) tanhf(x)
#endif
#else
#define TANHF(x) tanhf(x)
#endif

__device__ __forceinline__ float wave_sum(float v) {
#pragma unroll
  for (int off = 16; off > 0; off >>= 1) v += __shfl_xor(v, off, 32);
  return v;
}

__device__ __forceinline__ v4f tanh4(v4f x) {
  v4f r;
  r.x = TANHF(x.x); r.y = TANHF(x.y); r.z = TANHF(x.z); r.w = TANHF(x.w);
  return r;
}

__device__ __forceinline__ float sinh_over_r(float r) {
  if (r < 1e-3f) return 1.0f + r * r * (1.0f / 6.0f);
  return sinhf(r) / r;
}

// V = (U @ W) scaled per-column by S; written back into U in-place.
// U: LDS 32x128 (stride US). WL: LDS 128x128 (stride WS). S: 128 floats in LDS.
__device__ __forceinline__ void tangent_gemm(float* Uw, const float* WL,
                                             const float* Ssc, int lane) {
  const int nc = lane & 15;   // A row / B,C,D column within tile
  const int kh = lane >> 4;   // K half (K 0,1 vs 2,3)
  v8f acc[2][8];
#pragma unroll
  for (int mt = 0; mt < 2; ++mt)
#pragma unroll
    for (int nt = 0; nt < 8; ++nt)
      acc[mt][nt] = (v8f){0.f, 0.f, 0.f, 0.f, 0.f, 0.f, 0.f, 0.f};

  for (int k0 = 0; k0 < H_N; k0 += 4) {
    const int kr = k0 + 2 * kh;
    // A fragments (ds_load_2addr_b64 pairs)
    v2f a0 = *(const v2f*)(Uw + nc * US + kr);
    v2f a1 = *(const v2f*)(Uw + (16 + nc) * US + kr);
    // All 8 B fragments up front -> one load clause, one wait per K-step.
    v2f bf[8];
#pragma unroll
    for (int nt = 0; nt < 8; ++nt) {
      bf[nt].x = WL[kr * WS + nt * 16 + nc];
      bf[nt].y = WL[(kr + 1) * WS + nt * 16 + nc];
    }
#pragma unroll
    for (int nt = 0; nt < 8; ++nt) {
      acc[0][nt] = __builtin_amdgcn_wmma_f32_16x16x4_f32(
          false, a0, false, bf[nt], (short)0, acc[0][nt], false, false);
      acc[1][nt] = __builtin_amdgcn_wmma_f32_16x16x4_f32(
          false, a1, false, bf[nt], (short)0, acc[1][nt], false, false);
    }
  }
  // D layout: VGPR r -> tile row (r + 8*kh), column nc. Scale by S[col].
#pragma unroll
  for (int mt = 0; mt < 2; ++mt) {
#pragma unroll
    for (int nt = 0; nt < 8; ++nt) {
      float sc = Ssc[nt * 16 + nc];
#pragma unroll
      for (int r = 0; r < 8; ++r)
        Uw[(mt * 16 + 8 * kh + r) * US + nt * 16 + nc] = acc[mt][nt][r] * sc;
    }
  }
}

// rhs(t, y): dy = MLP([y,t]) per-lane component; div = trace of Jacobian.
__device__ __forceinline__ void eval_rhs(
    float t, float yv, int lane,
    const float* __restrict__ W0, const float* __restrict__ b0,
    const float* __restrict__ b1, const float* __restrict__ b2,
    const float* __restrict__ b3,
    const float* WL1, const float* WL2, const float* W3L,
    float* Uw, float* G, float* S1, float* S2, float* S3,
    float& dy, float& div) {
  // ---- primal layer 0: h1 = [y,t] @ W0 + b0  (W0 from global/L2)
  v4f h = *(const v4f*)(b0 + 4 * lane);
#pragma unroll 8
  for (int j = 0; j < D_N; ++j) {
    float yj = __shfl(yv, j, 32);
    v4f w = *(const v4f*)(W0 + j * H_N + 4 * lane);
    h += yj * w;
  }
  {
    v4f w = *(const v4f*)(W0 + D_N * H_N + 4 * lane);  // time row (row 32)
    h += t * w;
  }
  v4f g = tanh4(h);
  *(v4f*)(G + 4 * lane) = g;
  *(v4f*)(S1 + 4 * lane) = 1.0f - g * g;

  // ---- primal layer 1 (weights from LDS)
  h = *(const v4f*)(b1 + 4 * lane);
#pragma unroll 4
  for (int j = 0; j < H_N; j += 4) {
    v4f g4 = *(const v4f*)(G + j);
    h += g4.x * *(const v4f*)(WL1 + (j + 0) * WS + 4 * lane);
    h += g4.y * *(const v4f*)(WL1 + (j + 1) * WS + 4 * lane);
    h += g4.z * *(const v4f*)(WL1 + (j + 2) * WS + 4 * lane);
    h += g4.w * *(const v4f*)(WL1 + (j + 3) * WS + 4 * lane);
  }
  g = tanh4(h);
  *(v4f*)(G + 4 * lane) = g;
  *(v4f*)(S2 + 4 * lane) = 1.0f - g * g;

  // ---- primal layer 2
  h = *(const v4f*)(b2 + 4 * lane);
#pragma unroll 4
  for (int j = 0; j < H_N; j += 4) {
    v4f g4 = *(const v4f*)(G + j);
    h += g4.x * *(const v4f*)(WL2 + (j + 0) * WS + 4 * lane);
    h += g4.y * *(const v4f*)(WL2 + (j + 1) * WS + 4 * lane);
    h += g4.z * *(const v4f*)(WL2 + (j + 2) * WS + 4 * lane);
    h += g4.w * *(const v4f*)(WL2 + (j + 3) * WS + 4 * lane);
  }
  g = tanh4(h);
  *(v4f*)(G + 4 * lane) = g;
  *(v4f*)(S3 + 4 * lane) = 1.0f - g * g;

  // ---- primal layer 3: dy[m], lane m  (W3 from LDS)
  float d = b3[lane];
#pragma unroll 4
  for (int j = 0; j < H_N; j += 4) {
    v4f g4 = *(const v4f*)(G + j);
    d += g4.x * W3L[(j + 0) * D_N + lane];
    d += g4.y * W3L[(j + 1) * D_N + lane];
    d += g4.z * W3L[(j + 2) * D_N + lane];
    d += g4.w * W3L[(j + 3) * D_N + lane];
  }
  dy = d;

  // ---- tangent seed: U1[i][k] = W0[i][k] * s1[k], lane handles row i = lane
#pragma unroll
  for (int k = 0; k < H_N; k += 4) {
    v4f w = *(const v4f*)(W0 + lane * H_N + k);
    v4f s = *(const v4f*)(S1 + k);
    *(v4f*)(Uw + lane * US + k) = w * s;
  }
  // ---- tangent GEMMs (WMMA, weights from LDS)
  tangent_gemm(Uw, WL1, S2, lane);
  tangent_gemm(Uw, WL2, S3, lane);

  // ---- trace: div = sum_i sum_k U3[i][k] * W3[k][i], lane i = lane
  float tr = 0.0f;
#pragma unroll 4
  for (int k = 0; k < H_N; k += 4) {
    v4f u = *(const v4f*)(Uw + lane * US + k);
    tr += u.x * W3L[(k + 0) * D_N + lane];
    tr += u.y * W3L[(k + 1) * D_N + lane];
    tr += u.z * W3L[(k + 2) * D_N + lane];
    tr += u.w * W3L[(k + 3) * D_N + lane];
  }
  div = wave_sum(tr);
}

extern "C" __global__ __launch_bounds__(WAVES * 32, 1)
void hcnf_kernel(const float* __restrict__ z,
                 const float* __restrict__ W0, const float* __restrict__ b0,
                 const float* __restrict__ W1, const float* __restrict__ b1,
                 const float* __restrict__ W2, const float* __restrict__ b2,
                 const float* __restrict__ W3, const float* __restrict__ b3,
                 float* __restrict__ zt, float* __restrict__ logp) {
  __shared__ float WL1[H_N * WS];          //  69632 B
  __shared__ float WL2[H_N * WS];          //  69632 B
  __shared__ float W3L[H_N * D_N];         //  16384 B
  __shared__ float Ush[WAVES][32 * US];    // 135168 B
  __shared__ float Gsh[WAVES][H_N];
  __shared__ float S1sh[WAVES][H_N];
  __shared__ float S2sh[WAVES][H_N];
  __shared__ float S3sh[WAVES][H_N];       // 4x 4096 B -> total 307200 B

  const int tid = threadIdx.x;
  const int lane = tid & 31;
  const int wid = tid >> 5;
  const int b = blockIdx.x * WAVES + wid;

  // ---- cooperative staging of weights into LDS (once per block)
  for (int i4 = tid; i4 < (H_N * H_N) / 4; i4 += WAVES * 32) {
    int r = (i4 * 4) >> 7;
    int c = (i4 * 4) & 127;
    *(v4f*)(WL1 + r * WS + c) = *(const v4f*)(W1 + i4 * 4);
    *(v4f*)(WL2 + r * WS + c) = *(const v4f*)(W2 + i4 * 4);
  }
  for (int i4 = tid; i4 < (H_N * D_N) / 4; i4 += WAVES * 32)
    *(v4f*)(W3L + i4 * 4) = *(const v4f*)(W3 + i4 * 4);
  __syncthreads();

  float* Uw = Ush[wid];
  float* G = Gsh[wid];
  float* S1 = S1sh[wid];
  float* S2 = S2sh[wid];
  float* S3 = S3sh[wid];

  // ---- Lorentz log map at origin
  float z0 = z[b * (D_N + 1)];
  float xv = z[b * (D_N + 1) + 1 + lane];
  float alpha = fmaxf(z0, 1.0f + 1e-7f);
  float sq = sqrtf(alpha * alpha - 1.0f);
  float rA = acoshf(alpha);
  float yv = (rA / sq) * xv;

  float r0 = sqrtf(wave_sum(yv * yv) + 1e-12f);
  float ld0 = (float)(D_N - 1) * logf(sinh_over_r(r0));

  float lp = 0.0f;
  for (int step = 0; step < NSTEPS; ++step) {
    float t = (float)step * DT_F;
    float a1, d1, a2, d2, a3, d3, a4, d4;
    eval_rhs(t, yv, lane, W0, b0, b1, b2, b3, WL1, WL2, W3L,
             Uw, G, S1, S2, S3, a1, d1);
    eval_rhs(t + 0.5f * DT_F, yv + 0.5f * DT_F * a1, lane, W0, b0, b1, b2, b3,
             WL1, WL2, W3L, Uw, G, S1, S2, S3, a2, d2);
    eval_rhs(t + 0.5f * DT_F, yv + 0.5f * DT_F * a2, lane, W0, b0, b1, b2, b3,
             WL1, WL2, W3L, Uw, G, S1, S2, S3, a3, d3);
    eval_rhs(t + DT_F, yv + DT_F * a3, lane, W0, b0, b1, b2, b3,
             WL1, WL2, W3L, Uw, G, S1, S2, S3, a4, d4);
    yv += (DT_F / 6.0f) * (a1 + 2.0f * a2 + 2.0f * a3 + a4);
    lp += (DT_F / 6.0f) * (-(d1 + 2.0f * d2 + 2.0f * d3 + d4));
  }

  // ---- exp map at origin + log-density
  float rT = sqrtf(wave_sum(yv * yv) + 1e-12f);
  float sor = sinh_over_r(rT);
  zt[b * (D_N + 1) + 1 + lane] = sor * yv;
  if (lane == 0) {
    zt[b * (D_N + 1)] = coshf(rT);
    logp[b] = -lp - (float)(D_N - 1) * logf(sor) + ld0;
  }
}

extern "C" void kernel_launch(void* const* d_in, const int* in_sizes, int n_in,
                              void* d_out, int out_size, void* d_ws,
                              size_t ws_size, hipStream_t stream) {
  (void)in_sizes; (void)n_in; (void)d_ws; (void)ws_size; (void)out_size;
  const float* z  = (const float*)d_in[0];
  const float* W0 = (const float*)d_in[1];
  const float* b0 = (const float*)d_in[2];
  const float* W1 = (const float*)d_in[3];
  const float* b1 = (const float*)d_in[4];
  const float* W2 = (const float*)d_in[5];
  const float* b2 = (const float*)d_in[6];
  const float* W3 = (const float*)d_in[7];
  const float* b3 = (const float*)d_in[8];
  float* zt = (float*)d_out;                 // [4096, 33]
  float* logp = zt + B_N * (D_N + 1);        // [4096]

  dim3 grid(B_N / WAVES), block(WAVES * 32);
  hipLaunchKernelGGL(hcnf_kernel, grid, block, 0, stream,
                     z, W0, b0, W1, b1, W2, b2, W3, b3, zt, logp);
}